// MultiresolutionHashEncoder_12713103196377
// MI455X (gfx1250) — compile-verified
//
#include <hip/hip_runtime.h>
#include <cmath>

// ---------------------------------------------------------------------------
// Multiresolution hash encoding (Instant-NGP style), MI455X / gfx1250.
//
// Mapping: one thread per (point, level); level = lane & 15 so each wave
// handles 2 points x 16 levels and writes two contiguous 128B output rows
// with coalesced 8-byte non-temporal stores. Tables (64 MB total) are kept
// L2-resident by streaming x/out with NT hints so the 128 MB output stream
// does not evict them. Each corner fetch is a single global_load_b64
// (float2), 8 loads in flight per thread.
// ---------------------------------------------------------------------------

typedef float f32x2 __attribute__((ext_vector_type(2)));

#define LVLS      16
#define LOG2_T    19
#define TBL_MASK  ((1u << LOG2_T) - 1u)
#define PRIME1    2654435761u
#define PRIME2    805459861u

struct ResTable { float r[LVLS]; };

__global__ __launch_bounds__(256)
void hashgrid_encode_kernel(const float* __restrict__ x,        // (B,3)
                            const float* __restrict__ tables,   // (L,2^19,2)
                            const float* __restrict__ box_min,  // (3,)
                            const float* __restrict__ box_max,  // (3,)
                            float* __restrict__ out,            // (B,32)
                            ResTable rt,
                            int B)
{
    const int tid   = blockIdx.x * blockDim.x + threadIdx.x;
    const int point = tid >> 4;
    const int level = tid & 15;
    if (point >= B) return;   // grid sized exactly; never fires for B=1M

    // --- cooperative x load: lanes 0..5 load the wave's 6 floats, broadcast ---
    const int lane = threadIdx.x & 31;
    const int wave_first_point = (tid & ~31) >> 4;   // point handled by lane 0
    float xl = 0.0f;
    if (lane < 6)
        xl = __builtin_nontemporal_load(x + (size_t)wave_first_point * 3 + lane);
    const int half = lane >> 4;                      // which of the 2 points
    const float px = __shfl(xl, half * 3 + 0, 32);
    const float py = __shfl(xl, half * 3 + 1, 32);
    const float pz = __shfl(xl, half * 3 + 2, 32);

    const float bn0 = box_min[0], bn1 = box_min[1], bn2 = box_min[2];
    const float bx0 = box_max[0], bx1 = box_max[1], bx2 = box_max[2];

    // --- per-lane resolution via cndmask tree over constant-index kernarg ---
    float v0 = (level & 1) ? rt.r[1]  : rt.r[0];
    float v1 = (level & 1) ? rt.r[3]  : rt.r[2];
    float v2 = (level & 1) ? rt.r[5]  : rt.r[4];
    float v3 = (level & 1) ? rt.r[7]  : rt.r[6];
    float v4 = (level & 1) ? rt.r[9]  : rt.r[8];
    float v5 = (level & 1) ? rt.r[11] : rt.r[10];
    float v6 = (level & 1) ? rt.r[13] : rt.r[12];
    float v7 = (level & 1) ? rt.r[15] : rt.r[14];
    float w0 = (level & 2) ? v1 : v0;
    float w1 = (level & 2) ? v3 : v2;
    float w2 = (level & 2) ? v5 : v4;
    float w3 = (level & 2) ? v7 : v6;
    float u0 = (level & 4) ? w1 : w0;
    float u1 = (level & 4) ? w3 : w2;
    const float res = (level & 8) ? u1 : u0;

    // --- cell, corner base, trilinear weights (mirrors reference math) ---
    const float g0 = (bx0 - bn0) / res;
    const float g1 = (bx1 - bn1) / res;
    const float g2 = (bx2 - bn2) / res;

    const float xc0 = fminf(fmaxf(px, bn0), bx0);
    const float xc1 = fminf(fmaxf(py, bn1), bx1);
    const float xc2 = fminf(fmaxf(pz, bn2), bx2);

    const float f0 = floorf((xc0 - bn0) / g0);
    const float f1 = floorf((xc1 - bn1) / g1);
    const float f2 = floorf((xc2 - bn2) / g2);

    const float wx = (px - (f0 * g0 + bn0)) / g0;
    const float wy = (py - (f1 * g1 + bn1)) / g1;
    const float wz = (pz - (f2 * g2 + bn2)) / g2;

    const unsigned int c0 = (unsigned int)(int)f0;
    const unsigned int c1 = (unsigned int)(int)f1;
    const unsigned int c2 = (unsigned int)(int)f2;

    // --- 8 corner hashes (prime for dim0 is 1) ---
    const unsigned int hx0 = c0,                hx1 = c0 + 1u;
    const unsigned int hy0 = c1 * PRIME1,       hy1 = (c1 + 1u) * PRIME1;
    const unsigned int hz0 = c2 * PRIME2,       hz1 = (c2 + 1u) * PRIME2;

    const unsigned int lvl_base = (unsigned int)level << LOG2_T;
    const f32x2* __restrict__ tb = (const f32x2*)tables;

    // corner index c = i*4 + j*2 + k  (i<->wx, j<->wy, k<->wz)
    const f32x2 e0 = tb[lvl_base + ((hx0 ^ hy0 ^ hz0) & TBL_MASK)];
    const f32x2 e1 = tb[lvl_base + ((hx0 ^ hy0 ^ hz1) & TBL_MASK)];
    const f32x2 e2 = tb[lvl_base + ((hx0 ^ hy1 ^ hz0) & TBL_MASK)];
    const f32x2 e3 = tb[lvl_base + ((hx0 ^ hy1 ^ hz1) & TBL_MASK)];
    const f32x2 e4 = tb[lvl_base + ((hx1 ^ hy0 ^ hz0) & TBL_MASK)];
    const f32x2 e5 = tb[lvl_base + ((hx1 ^ hy0 ^ hz1) & TBL_MASK)];
    const f32x2 e6 = tb[lvl_base + ((hx1 ^ hy1 ^ hz0) & TBL_MASK)];
    const f32x2 e7 = tb[lvl_base + ((hx1 ^ hy1 ^ hz1) & TBL_MASK)];

    const float omx = 1.0f - wx, omy = 1.0f - wy, omz = 1.0f - wz;
    const f32x2 i00 = e0 * omx + e4 * wx;
    const f32x2 i01 = e1 * omx + e5 * wx;
    const f32x2 i10 = e2 * omx + e6 * wx;
    const f32x2 i11 = e3 * omx + e7 * wx;
    const f32x2 j0  = i00 * omy + i10 * wy;
    const f32x2 j1  = i01 * omy + i11 * wy;
    const f32x2 r   = j0 * omz + j1 * wz;

    // coalesced per-wave 2x128B contiguous NT store (keeps tables in L2)
    __builtin_nontemporal_store(r, (f32x2*)(out + (size_t)point * 32 + level * 2));
}

extern "C" void kernel_launch(void* const* d_in, const int* in_sizes, int n_in,
                              void* d_out, int out_size, void* d_ws, size_t ws_size,
                              hipStream_t stream) {
    const float* x       = (const float*)d_in[0];
    const float* tables  = (const float*)d_in[1];
    const float* box_min = (const float*)d_in[2];
    const float* box_max = (const float*)d_in[3];
    float* out = (float*)d_out;

    const int B = in_sizes[0] / 3;

    // Replicate numpy's float64 resolution table bit-for-bit via host libm
    // (floor(16 * b^l) sits within ~1e-12 of exact powers of two at
    //  l = 3,6,9,12,15 — hardcoding would be a coin flip).
    ResTable rt;
    const double b = std::exp((std::log(512.0) - std::log(16.0)) / 15.0);
    for (int l = 0; l < LVLS; ++l)
        rt.r[l] = (float)std::floor(16.0 * std::pow(b, (double)l));

    const int total   = B * LVLS;            // 16,777,216 for B = 1M
    const int block   = 256;                 // 8 wave32 per block
    const int nblocks = (total + block - 1) / block;

    hashgrid_encode_kernel<<<nblocks, block, 0, stream>>>(
        x, tables, box_min, box_max, out, rt, B);
}